// Softmax_82884278879162
// MI455X (gfx1250) — compile-verified
//
#include <hip/hip_runtime.h>
#include <cstdint>

// Quantized softmax, x: (B=8, R=4096, C=1024) fp32, softmax over R.
// Memory-bound (>=256MiB HBM traffic -> ~11us floor @ 23.3 TB/s); WMMA is
// irrelevant at ~0.1 FLOP/byte. CDNA5 features used instead:
//   * Tensor Data Mover (tensor_load_to_lds) descriptor-driven global->LDS DMA,
//     double-buffered per wave with TENSORcnt (s_wait_tensorcnt)
//   * non-temporal output stores (keep x resident in the 192MB L2 for passes 2/4)
//   * exact float-bits atomic-max (deterministic global quant scales)

#define B 8
#define R 4096
#define C 1024
#define CT 32            // columns owned per block
#define TILE_R 128       // rows per LDS tile (tile = 128*32*4B = 16KB, x2 buffers)
#define NT (R / TILE_R)  // 32 tiles per column set
#define NBLK (B * (C / CT))  // 256 blocks
#define NTHR 256
#define SLABS (NTHR / CT)    // 8 row-slabs per block
#define ROWS_W (TILE_R / 8)  // 16 rows DMA'd per wave per tile (2KB descriptor)
#define EPSQ 1e-9f

typedef uint32_t u32x4 __attribute__((ext_vector_type(4)));
typedef uint32_t u32x8 __attribute__((ext_vector_type(8)));

// fake-quant to a symmetric grid: clip(round(x/s), lo, hi)*s  (rintf == RTE == jnp.round)
__device__ __forceinline__ float fq_grid(float x, float s, float lo, float hi) {
  return fminf(fmaxf(rintf(x / s), lo), hi) * s;
}

// One TDM descriptor: rows x CT fp32 tile, row stride C elements, global->LDS.
// D# layout per CDNA5 ISA (2D: groups 0+1 only).
__device__ __forceinline__ void tdm_load_tile(uint64_t ga, uint32_t lds_byte, uint32_t rows) {
  u32x4 g0;
  u32x8 g1;
  g0[0] = 1u;                                 // count=1 (valid user descriptor)
  g0[1] = lds_byte;                           // lds_addr (bytes)
  g0[2] = (uint32_t)ga;                       // global_addr[31:0]
  g0[3] = (uint32_t)(ga >> 32) | (2u << 30);  // global_addr[56:32] | type=2 ("image")
  g1[0] = 2u << 16;                           // data_size=2 -> 4B; no multicast/pad/iterate
  g1[1] = (uint32_t)CT << 16;                 // tensor_dim0[15:0] = CT   (bits 79:48)
  g1[2] = rows << 16;                         // tensor_dim1[15:0] = rows (bits 111:80)
  g1[3] = (uint32_t)CT << 16;                 // tile_dim0 = CT           (bits 127:112)
  g1[4] = rows;                               // tile_dim1 = rows; tile_dim2 = 0
  g1[5] = (uint32_t)C;                        // tensor_dim0_stride = C elements (48b, low 32)
  g1[6] = 0u;                                 // stride hi / tensor_dim1_stride lo
  g1[7] = 0u;                                 // tensor_dim1_stride hi
  asm volatile("tensor_load_to_lds %0, %1" :: "s"(g0), "s"(g1) : "memory");
}

__global__ void k_init(float* gsub) { *(int*)gsub = 0; }  // 0x0 == +0.0f

// Pass 1: per-column max & min of xq (= fq(x, scale, int8)); derive global
// max|clip(xq-m,-12,0)| = max over columns of min(12, colmax-colmin).
__global__ void __launch_bounds__(NTHR) k_colstat(const float* __restrict__ x,
                                                  const float* __restrict__ scale_p,
                                                  float* __restrict__ m_out,
                                                  float* __restrict__ gsub) {
  __shared__ float tile[2][TILE_R * CT];
  __shared__ float rmx[SLABS][CT];
  __shared__ float rmn[SLABS][CT];
  const int tid = threadIdx.x;
  const int b   = blockIdx.x >> 5;          // C/CT == 32
  const int c0  = (blockIdx.x & 31) * CT;
  const float* base_b = x + (size_t)b * R * C;
  const uint32_t l0 = (uint32_t)(uintptr_t)&tile[0][0];
  const uint32_t l1 = (uint32_t)(uintptr_t)&tile[1][0];
  const float sc = scale_p[0];
  const int col = tid & (CT - 1), slab = tid >> 5;
  // wave id as a compiler-known scalar: each wave DMAs its own 16-row slice
  const uint32_t wid  = (uint32_t)__builtin_amdgcn_readfirstlane(tid >> 5);
  const uint32_t wlds = wid * (ROWS_W * CT * 4);
  const uint64_t gb   = (uint64_t)(uintptr_t)base_b + ((uint64_t)wid * ROWS_W * C + c0) * 4u;
  float mx = -3.4e38f, mn = 3.4e38f;

  tdm_load_tile(gb, l0 + wlds, ROWS_W);
  for (int t = 0; t < NT; ++t) {
    if (t + 1 < NT) {
      tdm_load_tile(gb + (uint64_t)(t + 1) * TILE_R * C * 4u,
                    (((t + 1) & 1) ? l1 : l0) + wlds, ROWS_W);
      __builtin_amdgcn_s_wait_tensorcnt(1);   // tile t done (TDM in-order per wave)
    } else {
      __builtin_amdgcn_s_wait_tensorcnt(0);
    }
    __syncthreads();                          // all waves' slices of tile t visible
    const float* tp = tile[t & 1];
    #pragma unroll 4
    for (int rr = slab; rr < TILE_R; rr += SLABS) {
      const float xq = fq_grid(tp[rr * CT + col], sc, -128.f, 127.f);
      mx = fmaxf(mx, xq);
      mn = fminf(mn, xq);
    }
    __syncthreads();                          // done reading before buffer is DMA'd again
  }
  rmx[slab][col] = mx; rmn[slab][col] = mn;
  __syncthreads();
  if (tid < CT) {
    float M = rmx[0][tid], N = rmn[0][tid];
    #pragma unroll
    for (int i = 1; i < SLABS; ++i) { M = fmaxf(M, rmx[i][tid]); N = fminf(N, rmn[i][tid]); }
    m_out[b * C + c0 + tid] = M;
    // candidate >= 0, so int-bit compare == float compare; max is exact -> deterministic
    atomicMax((int*)gsub, __float_as_int(fminf(12.0f, M - N)));
  }
}

__device__ __forceinline__ float eq_of(float v, float sc, float mcol, float s1) {
  const float xq  = fq_grid(v, sc, -128.f, 127.f);
  const float sub = fminf(fmaxf(xq - mcol, -12.f), 0.f);
  const float sq  = fq_grid(sub, s1, -32768.f, 32767.f);
  const float e   = __expf(sq);                 // TRANS op; e in (0,1]
  // global max(e) == 1 exactly (column max element gives sub==0 -> e==1),
  // so the e-quant scale is the constant 1/32767.
  return fq_grid(e, 1.f / 32767.f, -32768.f, 32767.f);
}

// Pass 2: per-column sum of e_q (fixed-order -> deterministic).
__global__ void __launch_bounds__(NTHR) k_expsum(const float* __restrict__ x,
                                                 const float* __restrict__ scale_p,
                                                 const float* __restrict__ m_in,
                                                 const float* __restrict__ gsub,
                                                 float* __restrict__ s_out) {
  __shared__ float tile[2][TILE_R * CT];
  __shared__ float rsum[SLABS][CT];
  const int tid = threadIdx.x;
  const int b   = blockIdx.x >> 5;
  const int c0  = (blockIdx.x & 31) * CT;
  const float* base_b = x + (size_t)b * R * C;
  const uint32_t l0 = (uint32_t)(uintptr_t)&tile[0][0];
  const uint32_t l1 = (uint32_t)(uintptr_t)&tile[1][0];
  const float sc = scale_p[0];
  const float s1 = fmaxf(gsub[0], EPSQ) / 32767.f;
  const int col = tid & (CT - 1), slab = tid >> 5;
  const float mcol = m_in[b * C + c0 + col];
  const uint32_t wid  = (uint32_t)__builtin_amdgcn_readfirstlane(tid >> 5);
  const uint32_t wlds = wid * (ROWS_W * CT * 4);
  const uint64_t gb   = (uint64_t)(uintptr_t)base_b + ((uint64_t)wid * ROWS_W * C + c0) * 4u;
  float acc = 0.f;

  tdm_load_tile(gb, l0 + wlds, ROWS_W);
  for (int t = 0; t < NT; ++t) {
    if (t + 1 < NT) {
      tdm_load_tile(gb + (uint64_t)(t + 1) * TILE_R * C * 4u,
                    (((t + 1) & 1) ? l1 : l0) + wlds, ROWS_W);
      __builtin_amdgcn_s_wait_tensorcnt(1);
    } else {
      __builtin_amdgcn_s_wait_tensorcnt(0);
    }
    __syncthreads();
    const float* tp = tile[t & 1];
    #pragma unroll 4
    for (int rr = slab; rr < TILE_R; rr += SLABS)
      acc += eq_of(tp[rr * CT + col], sc, mcol, s1);
    __syncthreads();
  }
  rsum[slab][col] = acc;
  __syncthreads();
  if (tid < CT) {
    float s = rsum[0][tid];
    #pragma unroll
    for (int i = 1; i < SLABS; ++i) s += rsum[i][tid];   // fixed order
    s_out[b * C + c0 + tid] = s;
  }
}

// Pass 3 (tiny): the chained scalar scales over the 8192 column sums.
__global__ void __launch_bounds__(1024) k_scales(const float* __restrict__ s_in,
                                                 float* __restrict__ rq_out,
                                                 float* __restrict__ s5_out) {
  __shared__ float red[1024];
  const int tid = threadIdx.x;
  float sv[8], rv[8];
  float mx = 0.f;
  #pragma unroll
  for (int k = 0; k < 8; ++k) { sv[k] = s_in[tid + k * 1024]; mx = fmaxf(mx, fabsf(sv[k])); }
  red[tid] = mx; __syncthreads();
  for (int o = 512; o > 0; o >>= 1) { if (tid < o) red[tid] = fmaxf(red[tid], red[tid + o]); __syncthreads(); }
  const float s3 = fmaxf(red[0], EPSQ) / 32767.f;   // sum scale
  __syncthreads();

  mx = 0.f;
  #pragma unroll
  for (int k = 0; k < 8; ++k) {
    const float sq = fq_grid(sv[k], s3, -32768.f, 32767.f);
    rv[k] = 1.f / sq;
    mx = fmaxf(mx, fabsf(rv[k]));
  }
  red[tid] = mx; __syncthreads();
  for (int o = 512; o > 0; o >>= 1) { if (tid < o) red[tid] = fmaxf(red[tid], red[tid + o]); __syncthreads(); }
  const float s4 = fmaxf(red[0], EPSQ) / 32767.f;   // reciprocal scale
  __syncthreads();

  mx = 0.f;
  #pragma unroll
  for (int k = 0; k < 8; ++k) {
    const float q = fq_grid(rv[k], s4, -32768.f, 32767.f);
    rq_out[tid + k * 1024] = q;
    mx = fmaxf(mx, fabsf(q));   // max|e_q*r_q| == max r_q since column-max e_q == 1
  }
  red[tid] = mx; __syncthreads();
  for (int o = 512; o > 0; o >>= 1) { if (tid < o) red[tid] = fmaxf(red[tid], red[tid + o]); __syncthreads(); }
  if (tid == 0) s5_out[0] = fmaxf(red[0], EPSQ) / 127.f;   // output (int8) scale
}

// Pass 4: out = fq(e_q * r_q, s5, int8); x read hits L2, output stored NT.
__global__ void __launch_bounds__(NTHR) k_finalize(const float* __restrict__ x,
                                                   const float* __restrict__ scale_p,
                                                   const float* __restrict__ m_in,
                                                   const float* __restrict__ gsub,
                                                   const float* __restrict__ rq_in,
                                                   const float* __restrict__ s5_in,
                                                   float* __restrict__ out) {
  __shared__ float tile[2][TILE_R * CT];
  const int tid = threadIdx.x;
  const int b   = blockIdx.x >> 5;
  const int c0  = (blockIdx.x & 31) * CT;
  const float* base_b = x + (size_t)b * R * C;
  const uint32_t l0 = (uint32_t)(uintptr_t)&tile[0][0];
  const uint32_t l1 = (uint32_t)(uintptr_t)&tile[1][0];
  const float sc = scale_p[0];
  const float s1 = fmaxf(gsub[0], EPSQ) / 32767.f;
  const float s5 = s5_in[0];
  const int col = tid & (CT - 1), slab = tid >> 5;
  const float mcol = m_in[b * C + c0 + col];
  const float rqc  = rq_in[b * C + c0 + col];
  const uint32_t wid  = (uint32_t)__builtin_amdgcn_readfirstlane(tid >> 5);
  const uint32_t wlds = wid * (ROWS_W * CT * 4);
  const uint64_t gb   = (uint64_t)(uintptr_t)base_b + ((uint64_t)wid * ROWS_W * C + c0) * 4u;

  tdm_load_tile(gb, l0 + wlds, ROWS_W);
  for (int t = 0; t < NT; ++t) {
    if (t + 1 < NT) {
      tdm_load_tile(gb + (uint64_t)(t + 1) * TILE_R * C * 4u,
                    (((t + 1) & 1) ? l1 : l0) + wlds, ROWS_W);
      __builtin_amdgcn_s_wait_tensorcnt(1);
    } else {
      __builtin_amdgcn_s_wait_tensorcnt(0);
    }
    __syncthreads();
    const float* tp = tile[t & 1];
    #pragma unroll 4
    for (int rr = slab; rr < TILE_R; rr += SLABS) {
      const float eq = eq_of(tp[rr * CT + col], sc, mcol, s1);
      const float oq = fq_grid(eq * rqc, s5, -128.f, 127.f);
      __builtin_nontemporal_store(oq, &out[((size_t)b * R + t * TILE_R + rr) * C + c0 + col]);
    }
    __syncthreads();
  }
}

extern "C" void kernel_launch(void* const* d_in, const int* in_sizes, int n_in,
                              void* d_out, int out_size, void* d_ws, size_t ws_size,
                              hipStream_t stream) {
  (void)in_sizes; (void)n_in; (void)out_size; (void)ws_size;
  const float* x       = (const float*)d_in[0];
  const float* scale_p = (const float*)d_in[1];
  float* out = (float*)d_out;

  // ws layout (floats): [0]=gsub, [1]=s5, [16..)=m(8192), then s(8192), then rq(8192). ~96KB.
  float* wsF  = (float*)d_ws;
  float* gsub = wsF + 0;
  float* s5   = wsF + 1;
  float* m    = wsF + 16;
  float* s    = wsF + 16 + 8192;
  float* rq   = wsF + 16 + 16384;

  k_init    <<<1,    1,    0, stream>>>(gsub);
  k_colstat <<<NBLK, NTHR, 0, stream>>>(x, scale_p, m, gsub);
  k_expsum  <<<NBLK, NTHR, 0, stream>>>(x, scale_p, m, gsub, s);
  k_scales  <<<1,    1024, 0, stream>>>(s, rq, s5);
  k_finalize<<<NBLK, NTHR, 0, stream>>>(x, scale_p, m, gsub, rq, s5, out);
}